// GaussianRenderer_59416577573119
// MI455X (gfx1250) — compile-verified
//
#include <hip/hip_runtime.h>
#include <math.h>
#include <stdint.h>

// ---------------- problem constants (match reference) ----------------
#define N_GAUSS  4096
#define IMG_H    128
#define IMG_W    128
#define HW       (IMG_H * IMG_W)
#define CHUNK    256            // gaussians staged in LDS per pass
#define NSEG     8              // depth segments rendered in parallel
#define SEGLEN   (N_GAUSS / NSEG)   // 512
#define NCHUNK_SEG (SEGLEN / CHUNK) // 2
#define NBATCH   (CHUNK / 16)   // 16-gaussian WMMA batches per chunk

#define ZNEAR_F   0.01f
#define ZFAR_F    100.0f
#define EPS2D_F   0.3f
#define ALPHA_MIN_F (1.0f / 255.0f)
#define ALPHA_MAX_F 0.999f

typedef float v2f __attribute__((ext_vector_type(2)));
typedef float v8f __attribute__((ext_vector_type(8)));

// =====================================================================
// Kernel 1: per-gaussian projection -> quadratic-form coefficients
//   wcoef[8]  (A-operand permuted slot order {w0,w1,w4,w5, w2,w3,0,0})
//   rgbop[4]  (r,g,b,opacity; opacity=0 when invalid)
//   keyf      (z when valid else +inf) for the depth sort
// Pixel feature vector: f = [px^2, py^2, px*py, px, py, 1, 0, 0]
// =====================================================================
__global__ void gs_preprocess(const float* __restrict__ Kmat,
                              const float* __restrict__ viewmat,
                              const float* __restrict__ means,
                              const float* __restrict__ quats,
                              const float* __restrict__ scales,
                              const float* __restrict__ opac,
                              const float* __restrict__ rgb,
                              float* __restrict__ wcoef,
                              float* __restrict__ rgbop,
                              float* __restrict__ keyf)
{
    int i = blockIdx.x * blockDim.x + threadIdx.x;
    if (i >= N_GAUSS) return;

    const float fx = Kmat[0], cx = Kmat[2], fy = Kmat[4], cy = Kmat[5];

    float R[3][3], t[3];
#pragma unroll
    for (int r = 0; r < 3; ++r) {
#pragma unroll
        for (int c = 0; c < 3; ++c) R[r][c] = viewmat[r * 4 + c];
        t[r] = viewmat[r * 4 + 3];
    }

    const float m0 = means[3 * i + 0], m1 = means[3 * i + 1], m2 = means[3 * i + 2];
    float p[3];
#pragma unroll
    for (int r = 0; r < 3; ++r)
        p[r] = R[r][0] * m0 + R[r][1] * m1 + R[r][2] * m2 + t[r];
    const float x = p[0], y = p[1], z = p[2];
    const float zs = (fabsf(z) < 1e-6f) ? 1e-6f : z;
    const float rz = 1.0f / zs;

    // quaternion -> rotation (normalized)
    float qw = quats[4 * i + 0], qx = quats[4 * i + 1],
          qy = quats[4 * i + 2], qz = quats[4 * i + 3];
    const float qn = rsqrtf(qw * qw + qx * qx + qy * qy + qz * qz);
    qw *= qn; qx *= qn; qy *= qn; qz *= qn;
    float Rg[3][3];
    Rg[0][0] = 1.f - 2.f * (qy * qy + qz * qz);
    Rg[0][1] = 2.f * (qx * qy - qw * qz);
    Rg[0][2] = 2.f * (qx * qz + qw * qy);
    Rg[1][0] = 2.f * (qx * qy + qw * qz);
    Rg[1][1] = 1.f - 2.f * (qx * qx + qz * qz);
    Rg[1][2] = 2.f * (qy * qz - qw * qx);
    Rg[2][0] = 2.f * (qx * qz - qw * qy);
    Rg[2][1] = 2.f * (qy * qz + qw * qx);
    Rg[2][2] = 1.f - 2.f * (qx * qx + qy * qy);

    const float s0 = scales[3 * i + 0], s1 = scales[3 * i + 1], s2 = scales[3 * i + 2];
    float M[3][3];
#pragma unroll
    for (int r = 0; r < 3; ++r) { M[r][0] = Rg[r][0] * s0; M[r][1] = Rg[r][1] * s1; M[r][2] = Rg[r][2] * s2; }
    float cov3d[3][3];
#pragma unroll
    for (int r = 0; r < 3; ++r)
#pragma unroll
        for (int c = 0; c < 3; ++c)
            cov3d[r][c] = M[r][0] * M[c][0] + M[r][1] * M[c][1] + M[r][2] * M[c][2];

    // covc = R * cov3d * R^T
    float T3[3][3];
#pragma unroll
    for (int r = 0; r < 3; ++r)
#pragma unroll
        for (int c = 0; c < 3; ++c)
            T3[r][c] = R[r][0] * cov3d[0][c] + R[r][1] * cov3d[1][c] + R[r][2] * cov3d[2][c];
    float covc[3][3];
#pragma unroll
    for (int r = 0; r < 3; ++r)
#pragma unroll
        for (int c = 0; c < 3; ++c)
            covc[r][c] = T3[r][0] * R[c][0] + T3[r][1] * R[c][1] + T3[r][2] * R[c][2];

    // J rows (sparse): J0 = (j00, 0, j02) ; J1 = (0, j11, j12)
    const float j00 = fx * rz, j02 = -fx * x * rz * rz;
    const float j11 = fy * rz, j12 = -fy * y * rz * rz;
    float t0[3], t1[3];
#pragma unroll
    for (int c = 0; c < 3; ++c) {
        t0[c] = j00 * covc[0][c] + j02 * covc[2][c];
        t1[c] = j11 * covc[1][c] + j12 * covc[2][c];
    }
    const float c00 = t0[0] * j00 + t0[2] * j02;
    const float c01 = t0[1] * j11 + t0[2] * j12;
    const float c11 = t1[1] * j11 + t1[2] * j12;

    const float a   = c00 + EPS2D_F;
    const float cc2 = c11 + EPS2D_F;
    const float b   = c01;
    const float det = a * cc2 - b * b;
    const float dets = (det > 0.f) ? det : 1.f;
    const float ica = cc2 / dets;     // ca
    const float icb = -b  / dets;     // cb
    const float icc = a   / dets;     // cc

    const float mx = fx * x * rz + cx;
    const float my = fy * y * rz + cy;
    const bool valid = (z > ZNEAR_F) && (z < ZFAR_F) && (det > 0.f);

    // quadratic-form coefficients: sigma = w . [px^2,py^2,px*py,px,py,1]
    const float w0 = 0.5f * ica;
    const float w1 = 0.5f * icc;
    const float w2 = icb;
    const float w3 = -(ica * mx + icb * my);
    const float w4 = -(icc * my + icb * mx);
    float       w5 = 0.5f * ica * mx * mx + 0.5f * icc * my * my + icb * mx * my;
    if (!valid) w5 = 1e10f;           // force sigma huge -> alpha 0

    // A-operand permuted slot order:
    // low lanes (K=0,1 / K=4,5) read slots 0..3 ; high lanes (K=2,3 / K=6,7) read slots 4..7
    float4* wc4 = (float4*)wcoef;
    wc4[i * 2 + 0] = make_float4(w0, w1, w4, w5);
    wc4[i * 2 + 1] = make_float4(w2, w3, 0.f, 0.f);

    float4* ro4 = (float4*)rgbop;
    ro4[i] = make_float4(rgb[3 * i + 0], rgb[3 * i + 1], rgb[3 * i + 2],
                         valid ? opac[i] : 0.f);

    keyf[i] = valid ? z : __builtin_inff();
}

// =====================================================================
// Kernel 2: single-workgroup LDS bitonic sort on (z_bits<<32 | idx),
// then gather coefficient / rgbop arrays into sorted order.
// =====================================================================
__global__ void gs_sort_gather(const float* __restrict__ keyf,
                               const float* __restrict__ wcoef,
                               const float* __restrict__ rgbop,
                               float* __restrict__ swcoef,
                               float* __restrict__ srgbop)
{
    __shared__ uint64_t k64[N_GAUSS];   // 32 KB
    const int t = threadIdx.x;
    const int NT = blockDim.x;

    for (int i = t; i < N_GAUSS; i += NT) {
        uint32_t fb = __float_as_uint(keyf[i]);
        k64[i] = ((uint64_t)fb << 32) | (uint32_t)i;
    }
    __syncthreads();

    for (int k = 2; k <= N_GAUSS; k <<= 1) {
        for (int j = k >> 1; j > 0; j >>= 1) {
            for (int i = t; i < N_GAUSS; i += NT) {
                int l = i ^ j;
                if (l > i) {
                    uint64_t va = k64[i], vb = k64[l];
                    bool up = ((i & k) == 0);
                    if ((va > vb) == up) { k64[i] = vb; k64[l] = va; }
                }
            }
            __syncthreads();
        }
    }

    const float4* wc4 = (const float4*)wcoef;
    const float4* ro4 = (const float4*)rgbop;
    float4* swc4 = (float4*)swcoef;
    float4* sro4 = (float4*)srgbop;
    for (int i = t; i < N_GAUSS; i += NT) {
        int src = (int)(k64[i] & 0xffffffffu);
        swc4[i * 2 + 0] = wc4[src * 2 + 0];
        swc4[i * 2 + 1] = wc4[src * 2 + 1];
        sro4[i]         = ro4[src];
    }
}

// =====================================================================
// Kernel 3: segmented render. grid = (IMG_H rows, NSEG depth segments),
// 256 threads = 8 waves, each wave owns 16 pixels. Each block composites
// its segment's SEGLEN sorted gaussians over one image row and emits a
// per-pixel partial (r,g,b,T). Sigma for each 16x16 tile = two chained
// V_WMMA_F32_16X16X4_F32 (K=8 quadratic form). D-layout gives each lane
// 8 consecutive-depth gaussians to composite sequentially in-register;
// half-waves merge with 4 shfl_xor(16): img = img1 + T1*img2, T = T1*T2.
// =====================================================================
__global__ void __launch_bounds__(256)
gs_render(const float* __restrict__ swcoef,
          const float* __restrict__ srgbop,
          float4* __restrict__ part)
{
    __shared__ float4 lds_w[CHUNK * 2];   // 8 KB  (coef slots)
    __shared__ float4 lds_ro[CHUNK];      // 4 KB  (r,g,b,op)

    const int tid  = threadIdx.x;
    const int lane = tid & 31;
    const int wave = tid >> 5;
    const bool hi  = (lane >= 16);
    const int col  = lane & 15;

    const int y    = blockIdx.x;
    const int seg  = blockIdx.y;
    const int gseg = seg * SEGLEN;
    const int pix  = wave * 16 + col;
    const float px = (float)pix + 0.5f;
    const float py = (float)y   + 0.5f;

    // B operand (pixel features), constant for the whole kernel.
    v2f b_lo, b_hi;
    if (!hi) { b_lo.x = px * px; b_lo.y = py * py; b_hi.x = py;  b_hi.y = 1.f; }
    else     { b_lo.x = px * py; b_lo.y = px;      b_hi.x = 0.f; b_hi.y = 0.f; }

    float rr = 0.f, rg = 0.f, rb = 0.f, rT = 1.f;

    const float4* swc4 = (const float4*)swcoef;
    const float4* sro4 = (const float4*)srgbop;

    for (int c = 0; c < NCHUNK_SEG; ++c) {
        __syncthreads();
        { // stage one gaussian per thread
            const int g = gseg + c * CHUNK + tid;
            lds_w[tid * 2 + 0] = swc4[g * 2 + 0];
            lds_w[tid * 2 + 1] = swc4[g * 2 + 1];
            lds_ro[tid]        = sro4[g];
        }
        __syncthreads();

        // pull next chunk toward the caches while this one computes
        if (c + 1 < NCHUNK_SEG) {
            const int gn = gseg + (c + 1) * CHUNK + tid;
            __builtin_prefetch(&swc4[gn * 2], 0, 0);
            __builtin_prefetch(&sro4[gn], 0, 0);
        }

        for (int bb = 0; bb < NBATCH; ++bb) {
            // A operand: gaussian (bb*16 + col), half-specific coef slots
            const float4 aw = lds_w[(bb * 16 + col) * 2 + (hi ? 1 : 0)];
            v2f a_lo, a_hi;
            a_lo.x = aw.x; a_lo.y = aw.y;
            a_hi.x = aw.z; a_hi.y = aw.w;

            v8f acc = {};
            acc = __builtin_amdgcn_wmma_f32_16x16x4_f32(
                      false, a_lo, false, b_lo, (short)0, acc, false, false);
            acc = __builtin_amdgcn_wmma_f32_16x16x4_f32(
                      false, a_hi, false, b_hi, (short)0, acc, false, false);

            // sequential front-to-back composite of this lane's 8 gaussians
            const int gb = bb * 16 + (hi ? 8 : 0);
            float pr = 0.f, pg = 0.f, pb = 0.f, pT = 1.f;
#pragma unroll
            for (int m = 0; m < 8; ++m) {
                const float4 ro = lds_ro[gb + m];
                const float sig = acc[m];
                float alpha = fminf(ALPHA_MAX_F, ro.w * __expf(-sig));
                alpha = (sig >= 0.f && alpha >= ALPHA_MIN_F) ? alpha : 0.f;
                const float wgt = alpha * pT;
                pr += wgt * ro.x;
                pg += wgt * ro.y;
                pb += wgt * ro.z;
                pT *= (1.f - alpha);
            }

            // merge half-batches (gaussians 0-7 then 8-15)
            const float qr = __shfl_xor(pr, 16, 32);
            const float qg = __shfl_xor(pg, 16, 32);
            const float qb = __shfl_xor(pb, 16, 32);
            const float qT = __shfl_xor(pT, 16, 32);
            float fr, fg, fb, fT;
            if (!hi) { fr = pr + pT * qr; fg = pg + pT * qg; fb = pb + pT * qb; fT = pT * qT; }
            else     { fr = qr + qT * pr; fg = qg + qT * pg; fb = qb + qT * pb; fT = qT * pT; }

            rr += rT * fr; rg += rT * fg; rb += rT * fb; rT *= fT;
        }
    }

    if (!hi) // one writer per pixel: segment partial (r,g,b,T)
        part[seg * HW + y * IMG_W + pix] = make_float4(rr, rg, rb, rT);
}

// =====================================================================
// Kernel 4: fold the NSEG depth-ordered segment partials per pixel.
// img = sum_s T_run * img_s ; T_run *= T_s   (bg is zero)
// Output layout (1, 3, H, W).
// =====================================================================
__global__ void gs_combine(const float4* __restrict__ part,
                           float* __restrict__ out)
{
    const int p = blockIdx.x * blockDim.x + threadIdx.x;
    if (p >= HW) return;
    float r = 0.f, g = 0.f, b = 0.f, T = 1.f;
#pragma unroll
    for (int s = 0; s < NSEG; ++s) {
        const float4 v = part[s * HW + p];
        r += T * v.x; g += T * v.y; b += T * v.z;
        T *= v.w;
    }
    out[0 * HW + p] = r;
    out[1 * HW + p] = g;
    out[2 * HW + p] = b;
}

// =====================================================================
extern "C" void kernel_launch(void* const* d_in, const int* in_sizes, int n_in,
                              void* d_out, int out_size, void* d_ws, size_t ws_size,
                              hipStream_t stream)
{
    const float* Ks       = (const float*)d_in[0]; // (1,3,3)
    const float* viewmats = (const float*)d_in[1]; // (1,4,4)
    const float* means    = (const float*)d_in[2]; // (N,3)
    const float* quats    = (const float*)d_in[3]; // (N,4)
    const float* scales   = (const float*)d_in[4]; // (N,3)
    const float* opac     = (const float*)d_in[5]; // (N)
    const float* rgb      = (const float*)d_in[6]; // (N,3)
    float* out = (float*)d_out;

    float* ws     = (float*)d_ws;
    float* wcoef  = ws;                       // N*8
    float* rgbop  = wcoef  + N_GAUSS * 8;     // N*4
    float* keyf   = rgbop  + N_GAUSS * 4;     // N
    float* swcoef = keyf   + N_GAUSS;         // N*8
    float* srgbop = swcoef + N_GAUSS * 8;     // N*4
    float4* part  = (float4*)(srgbop + N_GAUSS * 4); // NSEG*HW float4 (~2 MB)

    gs_preprocess<<<N_GAUSS / 256, 256, 0, stream>>>(
        Ks, viewmats, means, quats, scales, opac, rgb, wcoef, rgbop, keyf);

    gs_sort_gather<<<1, 1024, 0, stream>>>(keyf, wcoef, rgbop, swcoef, srgbop);

    gs_render<<<dim3(IMG_H, NSEG), 256, 0, stream>>>(swcoef, srgbop, part);

    gs_combine<<<(HW + 255) / 256, 256, 0, stream>>>(part, out);
}